// SoliditySyntaxLoss_807453852038
// MI455X (gfx1250) — compile-verified
//
#include <hip/hip_runtime.h>
#include <math.h>

// ---------------- problem constants ----------------
#define VSZ     50257          // vocab
#define NV4     12564          // VSZ / 4 (full float4 chunks, covers elements 0..50255)
#define SSEQ    1024
#define ROWS    2048           // B*S
#define T1      512            // threads per row-block (16 wave32)
#define NITER   25             // ceil(NV4 / T1)
#define DEPTH   8              // ring-buffer chunks in flight per thread

// LDS layout (floats): ring | tail | target | (m,s) reduction scratch
#define RING_FLOATS (T1 * 4 * DEPTH)        // 16384 floats = 64 KB
#define TAIL_IDX    RING_FLOATS             // element 50256 lands here
#define TGT_IDX     (TAIL_IDX + 1)          // extracted target logit
#define REDM_IDX    (TAIL_IDX + 4)
#define REDS_IDX    (REDM_IDX + 16)
#define SMEM_FLOATS (REDS_IDX + 16)         // 16420 floats ~ 65.7 KB -> 4 blocks/WGP

#define BIG_NEG (-3.0e38f)

// Low 32 bits of a generic pointer to shared memory == LDS byte address.
__device__ __forceinline__ unsigned lds_addr32(const void* p) {
    return (unsigned)(unsigned long long)(uintptr_t)p;
}

// ---------------- kernel 1: per-row CE, online softmax streamed through an LDS ring ----------------
__global__ __launch_bounds__(T1)
void ce_row_kernel(const float* __restrict__ logits,
                   const int*   __restrict__ tgt,
                   float*       __restrict__ row_loss) {
    extern __shared__ float lds[];
    const int tid  = threadIdx.x;
    const int lane = tid & 31;
    const int wid  = tid >> 5;

    const unsigned long long base =
        (unsigned long long)(logits + (size_t)blockIdx.x * VSZ);
    const unsigned ldsBase = lds_addr32(&lds[0]);
    const int t = tgt[blockIdx.x];          // 0..2047, scalar load

#define ISSUE_B128(slot, chunk)                                                  \
    do {                                                                         \
        unsigned go_ = (unsigned)(chunk) * 16u;                                  \
        unsigned lo_ = ldsBase + ((unsigned)(slot) * (unsigned)T1 +              \
                                  (unsigned)tid) * 16u;                          \
        asm volatile("global_load_async_to_lds_b128 %0, %1, %2"                  \
                     :: "v"(lo_), "v"(go_), "s"(base) : "memory");               \
    } while (0)

    // ---- prologue: fill the ring (chunks 0..DEPTH-1), wave-uniform issue count ----
#pragma unroll
    for (int i = 0; i < DEPTH; ++i) {
        int c = tid + i * T1;                       // < NV4 always for i < 8
        ISSUE_B128(i, c);
    }

    const float4* ring4 = (const float4*)lds;
    float m = BIG_NEG;
    float s = 0.f;

    // ---- main loop: wait -> consume chunk i (online m,s) -> refill slot ----
    // Per-wave async completion is IN ORDER; every thread issued the same count,
    // so: iters 0..17 wait ASYNCcnt<=7 (DEPTH-1); drain ladder 6..0 afterwards.
#pragma unroll
    for (int i = 0; i < NITER; ++i) {
        asm volatile("s_wait_asynccnt %0" :: "i"((i <= 17) ? (DEPTH - 1) : (24 - i))
                     : "memory");
        const int c = tid + i * T1;
        const int slot = i & (DEPTH - 1);
        if (c < NV4) {
            float4 v = ring4[slot * T1 + tid];
            if (i == 0 && tid == (t >> 2)) {        // target always in iteration 0
                lds[TGT_IDX] = ((t & 3) == 0) ? v.x :
                               ((t & 3) == 1) ? v.y :
                               ((t & 3) == 2) ? v.z : v.w;
            }
            float cm = fmaxf(fmaxf(v.x, v.y), fmaxf(v.z, v.w));
            float nm = fmaxf(m, cm);
            s = s * __expf(m - nm)
              + __expf(v.x - nm) + __expf(v.y - nm)
              + __expf(v.z - nm) + __expf(v.w - nm);
            m = nm;
        }
        if (i <= 16) {                              // refill: chunk i+DEPTH into this slot
            int cn = tid + (i + DEPTH) * T1;
            cn = (cn < NV4) ? cn : (NV4 - 1);       // clamp keeps counts wave-uniform
            asm volatile("s_wait_dscnt 0" ::: "memory");   // slot drained before overwrite
            ISSUE_B128(slot, cn);
        }
    }

    // ---- tail element 50256: uniform b32 issue by every thread, same destination ----
    {
        unsigned go = (unsigned)NV4 * 16u;
        unsigned lo = ldsBase + (unsigned)TAIL_IDX * 4u;
        asm volatile("global_load_async_to_lds_b32 %0, %1, %2"
                     :: "v"(lo), "v"(go), "s"(base) : "memory");
    }
    asm volatile("s_wait_asynccnt 0" ::: "memory");
    if (tid == 0) {
        float x  = lds[TAIL_IDX];
        float nm = fmaxf(m, x);
        s = s * __expf(m - nm) + __expf(x - nm);
        m = nm;
    }

    // ---- block-wide (m,s) logsumexp merge: wave32 shuffle + 16-wave LDS tree ----
    for (int o = 16; o > 0; o >>= 1) {
        float om = __shfl_xor(m, o, 32);
        float os = __shfl_xor(s, o, 32);
        float nm = fmaxf(m, om);
        s = s * __expf(m - nm) + os * __expf(om - nm);
        m = nm;
    }
    if (lane == 0) { lds[REDM_IDX + wid] = m; lds[REDS_IDX + wid] = s; }
    __syncthreads();
    if (wid == 0) {
        float mm = (lane < (T1 / 32)) ? lds[REDM_IDX + lane] : BIG_NEG;
        float ss = (lane < (T1 / 32)) ? lds[REDS_IDX + lane] : 0.f;
        for (int o = 8; o > 0; o >>= 1) {
            float om = __shfl_xor(mm, o, 32);
            float os = __shfl_xor(ss, o, 32);
            float nm = fmaxf(mm, om);
            ss = ss * __expf(mm - nm) + os * __expf(om - nm);
            mm = nm;
        }
        if (lane == 0) {
            float xt = lds[TGT_IDX];
            row_loss[blockIdx.x] = (mm + logf(ss)) - xt;   // -log_softmax[target]
        }
    }
#undef ISSUE_B128
}

// ---------------- kernel 2: deterministic finalize (CE mean + syntax penalty) ----------------
__global__ __launch_bounds__(256)
void finalize_kernel(const int*   __restrict__ tg,
                     const float* __restrict__ row_loss,
                     float*       __restrict__ out) {
    __shared__ float sred[8 * 5];
    const int tid  = threadIdx.x;
    const int lane = tid & 31;
    const int wid  = tid >> 5;

    float ce = 0.f, nkw = 0.f, nrbc = 0.f, npar = 0.f, nbrc = 0.f;

    for (int r = tid; r < ROWS; r += 256) ce += row_loss[r];

    // follower sets as 32-bit masks (all follower ids < 32)
    const unsigned FM[19] = {
        (1u<<10)|(1u<<17)|(1u<<18)|(1u<<19)|(1u<<20)|(1u<<21)|(1u<<22), // 1000
        (1u<<12)|(1u<<23)|(1u<<24),                                    // 1001
        (1u<<10),(1u<<10),(1u<<10),(1u<<10),(1u<<10),(1u<<10),(1u<<10),// 1002-1008
        (1u<<14)|(1u<<10),                                             // 1009
        (1u<<14),                                                      // 1010
        (1u<<14),                                                      // 1011
        (1u<<15)|(1u<<16),                                             // 1012
        (1u<<25),                                                      // 1013
        (1u<<12),(1u<<12),                                             // 1014-1015
        (1u<<10),                                                      // 1016
        (1u<<12)|(1u<<10),                                             // 1017
        (1u<<10)                                                       // 1018
    };

    for (int p = tid; p < 2 * (SSEQ - 1); p += 256) {
        const int  b = p / (SSEQ - 1);
        const int  i = p % (SSEQ - 1);
        const int* tp = tg + b * SSEQ;
        const int cur = tp[i];
        const int nxt = tp[i + 1];

        if (cur >= 1000 && cur <= 1018) {
            bool ok = (nxt >= 0) && (nxt < 32) && ((FM[cur - 1000] >> nxt) & 1u);
            if (!ok) nkw += 1.f;
        }
        if ((cur == 1009 || cur == 1010 || cur == 1011) && nxt != 14) nrbc += 1.f;
        if (cur == 10) {   // open paren: CLOSE_PAREN within next 19 tokens
            int end = min(i + 19, SSEQ - 1);
            bool f = false;
            for (int j = i + 1; j <= end; ++j) if (tp[j] == 11) { f = true; break; }
            if (!f) npar += 1.f;
        }
        if (cur == 12) {   // open brace: CLOSE_BRACE within next 49 tokens
            int end = min(i + 49, SSEQ - 1);
            bool f = false;
            for (int j = i + 1; j <= end; ++j) if (tp[j] == 13) { f = true; break; }
            if (!f) nbrc += 1.f;
        }
    }

    for (int o = 16; o > 0; o >>= 1) {
        ce   += __shfl_xor(ce,   o, 32);
        nkw  += __shfl_xor(nkw,  o, 32);
        nrbc += __shfl_xor(nrbc, o, 32);
        npar += __shfl_xor(npar, o, 32);
        nbrc += __shfl_xor(nbrc, o, 32);
    }
    if (lane == 0) {
        sred[wid]      = ce;
        sred[8  + wid] = nkw;
        sred[16 + wid] = nrbc;
        sred[24 + wid] = npar;
        sred[32 + wid] = nbrc;
    }
    __syncthreads();
    if (tid == 0) {
        float tce = 0, tkw = 0, trbc = 0, tpar = 0, tbrc = 0;
        for (int w = 0; w < 8; ++w) {
            tce  += sred[w];       tkw  += sred[8 + w];  trbc += sred[16 + w];
            tpar += sred[24 + w];  tbrc += sred[32 + w];
        }
        float count = tkw + trbc + tpar + tbrc;
        float total = 2.0f * tkw + 1.5f * trbc + tpar + tbrc;
        float pen = (count > 0.f) ? (total / fmaxf(count, 1.f)) : 0.f;
        out[0] = tce / (float)ROWS + 0.1f * pen;
    }
}

// ---------------- host launcher ----------------
extern "C" void kernel_launch(void* const* d_in, const int* in_sizes, int n_in,
                              void* d_out, int out_size, void* d_ws, size_t ws_size,
                              hipStream_t stream) {
    const float* logits = (const float*)d_in[0];   // (2,1024,50257) f32
    const int*   tgt    = (const int*)d_in[1];     // (2,1024) i32
    float* out = (float*)d_out;                    // scalar f32
    float* ws  = (float*)d_ws;                     // 2048 row losses

    const size_t smem = (size_t)SMEM_FLOATS * sizeof(float);  // ~65.7 KB -> 4 blocks/WGP
    ce_row_kernel<<<ROWS, T1, smem, stream>>>(logits, tgt, ws);
    finalize_kernel<<<1, 256, 0, stream>>>(tgt, ws, out);
}